// SPN_76330158784729
// MI455X (gfx1250) — compile-verified
//
#include <hip/hip_runtime.h>
#include <math.h>

// Problem constants from the reference
#define NF    32
#define BATCH 4
#define HH    256
#define WW    256

typedef __attribute__((ext_vector_type(2))) float v2f;
typedef __attribute__((ext_vector_type(8))) float v8f;

// ---------------------------------------------------------------------------
// Weight repack: OIHW -> [ocTile][k][16] with k = tap*Cp + cin, Cp = Cin
// rounded up to a multiple of 4 (zero padded), oc padded to tile of 16.
// Makes conv A-fragment loads two coalesced 64B loads with no guards and
// removes all div/mod from the conv inner loop.
// ---------------------------------------------------------------------------
__global__ void repack_weights_kernel(const float* __restrict__ w,
                                      float* __restrict__ wT,
                                      int Cin, int Cout, int KS)
{
  const int Cp    = (Cin + 3) & ~3;
  const int Kp    = KS * KS * Cp;
  const int tiles = (Cout + 15) >> 4;
  const size_t total = (size_t)tiles * Kp * 16;
  const size_t n = (size_t)blockIdx.x * blockDim.x + threadIdx.x;
  if (n >= total) return;
  const int m    = (int)(n & 15);
  size_t r       = n >> 4;
  const int k    = (int)(r % Kp);
  const int tile = (int)(r / Kp);
  const int tap  = k / Cp;
  const int cin  = k - tap * Cp;
  const int oc   = tile * 16 + m;
  float v = 0.f;
  if (oc < Cout && cin < Cin)
    v = w[((size_t)oc * Cin + cin) * (KS * KS) + tap];
  wT[n] = v;
}

// ---------------------------------------------------------------------------
// Implicit-GEMM convolution on V_WMMA_F32_16X16X4_F32 (fp32 matrix pipe).
//
// One wave computes 16 output channels x 64 pixels (4 N-tiles of 16 along x,
// 4 accumulators sharing one A fragment). K is consumed as
// (tap outer, 4 input channels inner):
//   per chunk: 2 coalesced A loads + 8 B loads + 4 WMMAs (8192 FLOP).
//
// fp32 WMMA VGPR layouts (CDNA5 ISA 7.12.2):
//   A (16x4):  lanes 0-15 -> M=lane, {K0,K1}; lanes 16-31 -> M=lane-16,{K2,K3}
//   B (4x16):  lanes 0-15 -> N=lane, {K0,K1}; lanes 16-31 -> N=lane-16,{K2,K3}
//   C/D:       VGPR r: lanes 0-15 -> (M=r, N=lane); 16-31 -> (M=r+8, N=lane-16)
//
// MODE 0: out = acc + bias      MODE 1: out = elu(acc + bias) + skip
// ---------------------------------------------------------------------------
template <int KS, int MODE>
__global__ void conv_wmma_kernel(const float* __restrict__ in,
                                 const float* __restrict__ wT,
                                 const float* __restrict__ bias,
                                 const float* __restrict__ skip,
                                 float* __restrict__ out,
                                 int Bn, int Cin, int Cout, int H, int W)
{
  const int lane = threadIdx.x & 31;
  const int wave = threadIdx.x >> 5;
  const int wpb  = blockDim.x >> 5;
  const int x0   = (blockIdx.x * wpb + wave) * 64;   // 4 tiles of 16 px
  if (x0 >= W) return;                                // uniform per wave
  int nT = (W - x0) >> 4; if (nT > 4) nT = 4;         // active tiles (uniform)

  const int y       = blockIdx.y;
  const int ocTiles = (Cout + 15) >> 4;
  const int b       = blockIdx.z / ocTiles;
  const int ot      = blockIdx.z % ocTiles;
  const int oc0     = ot * 16;

  const int kh  = lane >> 4;    // 0 -> K slots {0,1}, 1 -> {2,3}
  const int idx = lane & 15;    // M for A-frag, N for B-frag
  const int Cp  = (Cin + 3) & ~3;
  const int pad = KS >> 1;
  const size_t HW = (size_t)H * W;

  const float* __restrict__ inB = in + (size_t)b * Cin * HW;
  const float* __restrict__ wTt = wT + (size_t)ot * (KS * KS * Cp) * 16;

  v8f acc[4];
#pragma unroll
  for (int t = 0; t < 4; ++t) acc[t] = (v8f){0.f,0.f,0.f,0.f,0.f,0.f,0.f,0.f};

  const bool cinPadded = (Cp != Cin);   // only the Cin=3 stem convs

#pragma unroll
  for (int ty = 0; ty < KS; ++ty) {
    const int yy = y + ty - pad;
    const bool rowOk = (unsigned)yy < (unsigned)H;
#pragma unroll
    for (int tx = 0; tx < KS; ++tx) {
      const int tap = ty * KS + tx;
      const float* __restrict__ wk = wTt + (size_t)tap * Cp * 16;

      // Per-tile base pointers + validity (hoisted out of the K loop)
      const float* bp[4];
      bool ok[4];
#pragma unroll
      for (int t = 0; t < 4; ++t) {
        const int xx = x0 + 16 * t + idx + tx - pad;
        ok[t] = rowOk && ((unsigned)xx < (unsigned)W);
        bp[t] = inB + (size_t)yy * W + xx;
      }

      // Hot loop: 4 input channels per trip -> 4 WMMAs
      for (int c0 = 0; c0 < Cp; c0 += 4) {
        const int kc = c0 + kh * 2;
        v2f afrag;
        afrag.x = wk[(kc + 0) * 16 + idx];   // coalesced, zero-padded
        afrag.y = wk[(kc + 1) * 16 + idx];
        const bool k0ok = !cinPadded || (kc + 0) < Cin;
        const bool k1ok = !cinPadded || (kc + 1) < Cin;
#pragma unroll
        for (int t = 0; t < 4; ++t) {
          if (t < nT) {
            v2f bfrag;
            bfrag.x = (ok[t] && k0ok) ? bp[t][(size_t)(kc + 0) * HW] : 0.f;
            bfrag.y = (ok[t] && k1ok) ? bp[t][(size_t)(kc + 1) * HW] : 0.f;
            acc[t] = __builtin_amdgcn_wmma_f32_16x16x4_f32(
                false, afrag, false, bfrag, (short)0, acc[t], false, false);
          }
        }
      }
    }
  }

  // Epilogue: bias (+ ELU + skip), coalesced 16-lane stores
#pragma unroll
  for (int t = 0; t < 4; ++t) {
    if (t < nT) {
      const int xN = x0 + 16 * t + idx;
#pragma unroll
      for (int r = 0; r < 8; ++r) {
        const int oc = oc0 + r + kh * 8;
        if (oc < Cout) {
          float v = acc[t][r] + bias[oc];
          const size_t addr = (((size_t)b * Cout + oc) * H + y) * W + xN;
          if (MODE == 1) {
            v = (v > 0.f) ? v : (__expf(v) - 1.f);   // ELU(alpha=1)
            v += skip[addr];
          }
          out[addr] = v;
        }
      }
    }
  }
}

// ---------------------------------------------------------------------------
// ReLU + 3x3/stride-2/pad-1 maxpool fused (relu monotone -> max then clamp)
// ---------------------------------------------------------------------------
__global__ void maxpool_relu_kernel(const float* __restrict__ in,
                                    float* __restrict__ out,
                                    int planes, int H, int W)
{
  const int Ho = H >> 1, Wo = W >> 1;
  const size_t total = (size_t)planes * Ho * Wo;
  const size_t n = (size_t)blockIdx.x * blockDim.x + threadIdx.x;
  if (n >= total) return;
  const int ox = (int)(n % Wo);
  size_t r = n / Wo;
  const int oy = (int)(r % Ho);
  const int pc = (int)(r / Ho);
  const float* __restrict__ p = in + (size_t)pc * H * W;
  float m = -INFINITY;
#pragma unroll
  for (int dy = -1; dy <= 1; ++dy) {
    const int yy = 2 * oy + dy;
    if (yy < 0 || yy >= H) continue;
#pragma unroll
    for (int dx = -1; dx <= 1; ++dx) {
      const int xx = 2 * ox + dx;
      if (xx < 0 || xx >= W) continue;
      m = fmaxf(m, p[yy * W + xx]);
    }
  }
  out[n] = fmaxf(m, 0.f);
}

// ---------------------------------------------------------------------------
// 2x bilinear upsample, half-pixel centers (jax.image.resize 'bilinear').
// in_coord = out*0.5 - 0.25, edge-clamped.
// ---------------------------------------------------------------------------
__global__ void up2_kernel(const float* __restrict__ in,
                           float* __restrict__ out,
                           int planes, int Hin, int Win)
{
  const int Ho = Hin * 2, Wo = Win * 2;
  const size_t total = (size_t)planes * Ho * Wo;
  const size_t n = (size_t)blockIdx.x * blockDim.x + threadIdx.x;
  if (n >= total) return;
  const int ox = (int)(n % Wo);
  size_t r = n / Wo;
  const int oy = (int)(r % Ho);
  const int pc = (int)(r / Ho);

  const float fy = oy * 0.5f - 0.25f;
  const float fx = ox * 0.5f - 0.25f;
  int y0 = (int)floorf(fy);  const float wy = fy - (float)y0;
  int x0 = (int)floorf(fx);  const float wx = fx - (float)x0;
  const int y0c = min(max(y0, 0), Hin - 1);
  const int y1c = min(max(y0 + 1, 0), Hin - 1);
  const int x0c = min(max(x0, 0), Win - 1);
  const int x1c = min(max(x0 + 1, 0), Win - 1);

  const float* __restrict__ p = in + (size_t)pc * Hin * Win;
  const float v00 = p[y0c * Win + x0c], v01 = p[y0c * Win + x1c];
  const float v10 = p[y1c * Win + x0c], v11 = p[y1c * Win + x1c];
  out[n] = (1.f - wy) * ((1.f - wx) * v00 + wx * v01)
         +        wy  * ((1.f - wx) * v10 + wx * v11);
}

// ---------------------------------------------------------------------------
// Gate normalization (in place). gates layout (B, 96, H, W):
// channels [0,32)=g1, [32,64)=g2, [64,96)=g3 for one direction.
// ---------------------------------------------------------------------------
__global__ void norm_gates_kernel(float* __restrict__ g)
{
  const size_t plane = (size_t)HH * WW;
  const size_t total = (size_t)BATCH * NF * plane;
  const size_t n = (size_t)blockIdx.x * blockDim.x + threadIdx.x;
  if (n >= total) return;
  const size_t s = n % plane;
  size_t r = n / plane;
  const int c = (int)(r % NF);
  const int b = (int)(r / NF);
  float* g1 = g + ((size_t)b * 96 + c) * plane + s;
  float* g2 = g1 + 32 * plane;
  float* g3 = g1 + 64 * plane;
  const float a1 = *g1, a2 = *g2, a3 = *g3;
  float sum = fabsf(a1) + fabsf(a2) + fabsf(a3);
  if (sum == 0.f) sum = 1e-6f;
  if (sum >= 1.f) {
    const float inv = 1.f / sum;
    *g1 = a1 * inv; *g2 = a2 * inv; *g3 = a3 * inv;
  }
}

// ---------------------------------------------------------------------------
// GateRecurrent2dnoind: sequential gated scan. One block per (b,c) plane;
// 256 threads each own one element of the 256-wide state vector (in LDS),
// marching 256 steps along the scan axis (tridiagonal exchange via LDS).
//   hor=1: scan over x (t), state over y (i), addr = i*W + t
//   hor=0: scan over y (t), state over x (i), addr = t*W + i
// Result folded into outMax (store for first direction, fmax after).
// ---------------------------------------------------------------------------
__global__ void propagate_kernel(const float* __restrict__ X,
                                 const float* __restrict__ gates,
                                 float* __restrict__ outMax,
                                 int hor, int rev, int first)
{
  __shared__ float hs[HH + 2];   // [0] and [HH+1] are zero boundaries
  const int c = blockIdx.x;
  const int b = blockIdx.y;
  const int i = threadIdx.x;     // state index 0..255
  const size_t plane = (size_t)HH * WW;

  const float* __restrict__ xP  = X     + ((size_t)b * NF + c) * plane;
  const float* __restrict__ g1P = gates + ((size_t)b * 96 + c) * plane;
  const float* __restrict__ g2P = g1P + 32 * plane;
  const float* __restrict__ g3P = g1P + 64 * plane;
  float* __restrict__ oP = outMax + ((size_t)b * NF + c) * plane;

  if (i == 0) { hs[0] = 0.f; hs[HH + 1] = 0.f; }
  hs[i + 1] = 0.f;             // h[-1] = 0
  __syncthreads();

  const int T = WW;            // H == W == 256
  for (int t = 0; t < T; ++t) {
    const int tt = rev ? (T - 1 - t) : t;
    const size_t addr = hor ? ((size_t)i * WW + tt) : ((size_t)tt * WW + i);
    const float xv = xP[addr];
    const float a1 = g1P[addr], a2 = g2P[addr], a3 = g3P[addr];
    const float hu = hs[i], hm = hs[i + 1], hd = hs[i + 2];
    const float hn = (1.f - a1 - a2 - a3) * xv + a1 * hu + a2 * hm + a3 * hd;
    __syncthreads();           // all reads of step t-1 state done
    hs[i + 1] = hn;
    __syncthreads();           // state for step t visible
    if (first) oP[addr] = hn;
    else       oP[addr] = fmaxf(oP[addr], hn);
  }
}

// ---------------------------------------------------------------------------
// Host-side launch plumbing
// ---------------------------------------------------------------------------
static inline int gblocks(size_t total) { return (int)((total + 255) / 256); }

static void launch_conv(hipStream_t s, const float* in, const float* w,
                        const float* bias, const float* skip, float* out,
                        float* wT, int Bn, int Cin, int Cout, int H, int W,
                        int KS, int mode)
{
  // 1) repack weights (cheap, bandwidth-trivial)
  const int Cp    = (Cin + 3) & ~3;
  const int Kp    = KS * KS * Cp;
  const int tiles = (Cout + 15) >> 4;
  const size_t wtotal = (size_t)tiles * Kp * 16;
  repack_weights_kernel<<<gblocks(wtotal), 256, 0, s>>>(w, wT, Cin, Cout, KS);

  // 2) conv: each wave covers 64 px along x
  const int tilesX = (W + 15) / 16;
  const int wavesX = (tilesX + 3) / 4;
  const int wpb    = wavesX < 4 ? wavesX : 4;
  dim3 blk(32 * wpb);
  dim3 grd((wavesX + wpb - 1) / wpb, H, Bn * tiles);
  if (KS == 3) {
    if (mode == 0)
      conv_wmma_kernel<3, 0><<<grd, blk, 0, s>>>(in, wT, bias, skip, out, Bn, Cin, Cout, H, W);
    else
      conv_wmma_kernel<3, 1><<<grd, blk, 0, s>>>(in, wT, bias, skip, out, Bn, Cin, Cout, H, W);
  } else {
    conv_wmma_kernel<1, 0><<<grd, blk, 0, s>>>(in, wT, bias, skip, out, Bn, Cin, Cout, H, W);
  }
}

extern "C" void kernel_launch(void* const* d_in, const int* in_sizes, int n_in,
                              void* d_out, int out_size, void* d_ws, size_t ws_size,
                              hipStream_t stream)
{
  const float* x   = (const float*)d_in[0];
  const float* rgb = (const float*)d_in[1];
  const float* mw  = (const float*)d_in[2];   const float* mb  = (const float*)d_in[3];
  const float* c1w = (const float*)d_in[4];   const float* c1b = (const float*)d_in[5];
  const float* c2w = (const float*)d_in[6];   const float* c2b = (const float*)d_in[7];
  const float* c3w = (const float*)d_in[8];   const float* c3b = (const float*)d_in[9];
  const float* c4w = (const float*)d_in[10];  const float* c4b = (const float*)d_in[11];
  const float* d0w = (const float*)d_in[12];  const float* d0b = (const float*)d_in[13];
  const float* d2w = (const float*)d_in[14];  const float* d2b = (const float*)d_in[15];
  const float* d4w = (const float*)d_in[16];  const float* d4b = (const float*)d_in[17];
  const float* d6w = (const float*)d_in[18];  const float* d6b = (const float*)d_in[19];
  const float* d7w = (const float*)d_in[20];  const float* d7b = (const float*)d_in[21];
  const float* pw  = (const float*)d_in[22];  const float* pb  = (const float*)d_in[23];

  float* ws = (float*)d_ws;

  // Workspace layout (floats). Total ~86.3M floats = ~345 MB.
  const size_t plane256 = (size_t)HH * WW;                     // 65536
  float* Xb     = ws;                                          // (4,32,256,256)  8.39M
  float* f1     = Xb     + (size_t)BATCH * 32  * plane256;     // (4,32,256,256)  8.39M
  float* f2     = f1     + (size_t)BATCH * 32  * plane256;     // (4,64,128,128)  4.19M
  float* f3     = f2     + (size_t)BATCH * 64  * 128 * 128;    // (4,128,64,64)   2.10M
  float* o3     = f3     + (size_t)BATCH * 128 * 64 * 64;      // (4,32,256,256)  8.39M
  float* gates  = o3     + (size_t)BATCH * 32  * plane256;     // (4,96,256,256) 25.17M
  float* maxacc = gates  + (size_t)BATCH * 96  * plane256;     // (4,32,256,256)  8.39M
  float* S1     = maxacc + (size_t)BATCH * 32  * plane256;     // 4.19M scratch
  float* S2     = S1     + (size_t)BATCH * 64  * 128 * 128;    // 16.78M scratch
  float* wT     = S2     + (size_t)BATCH * 64  * plane256;     // 0.30M repacked weights

  // ------------------ encoder ------------------
  launch_conv(stream, x,   mw,  mb,  nullptr, Xb, wT, BATCH, 3,   32,  256, 256, 3, 0);
  launch_conv(stream, rgb, c1w, c1b, nullptr, f1, wT, BATCH, 3,   32,  256, 256, 3, 0);
  maxpool_relu_kernel<<<gblocks((size_t)BATCH*32*128*128), 256, 0, stream>>>(f1, S1, BATCH*32, 256, 256);
  launch_conv(stream, S1, c2w, c2b, nullptr, f2, wT, BATCH, 32,  64,  128, 128, 3, 0);
  maxpool_relu_kernel<<<gblocks((size_t)BATCH*64*64*64), 256, 0, stream>>>(f2, S1, BATCH*64, 128, 128);
  launch_conv(stream, S1, c3w, c3b, nullptr, f3, wT, BATCH, 64,  128, 64,  64, 3, 0);
  maxpool_relu_kernel<<<gblocks((size_t)BATCH*128*32*32), 256, 0, stream>>>(f3, S1, BATCH*128, 64, 64);
  launch_conv(stream, S1, c4w, c4b, nullptr, S2, wT, BATCH, 128, 256, 32,  32, 3, 0);

  // ------------------ decoder ------------------
  launch_conv(stream, S2, d0w, d0b, nullptr, S1, wT, BATCH, 256, 128, 32, 32, 1, 0);
  up2_kernel<<<gblocks((size_t)BATCH*128*64*64), 256, 0, stream>>>(S1, S2, BATCH*128, 32, 32);
  launch_conv(stream, S2, d2w, d2b, f3, S1, wT, BATCH, 128, 128, 64, 64, 3, 1);
  up2_kernel<<<gblocks((size_t)BATCH*128*128*128), 256, 0, stream>>>(S1, S2, BATCH*128, 64, 64);
  launch_conv(stream, S2, d4w, d4b, f2, S1, wT, BATCH, 128, 64, 128, 128, 3, 1);
  up2_kernel<<<gblocks((size_t)BATCH*64*256*256), 256, 0, stream>>>(S1, S2, BATCH*64, 128, 128);
  launch_conv(stream, S2, d6w, d6b, f1, o3, wT, BATCH, 64, 32, 256, 256, 3, 1);

  // ------------------ guide + 4 directional scans (96 ch/direction) --------
  const int dirs_hor[4] = {1, 1, 0, 0};
  const int dirs_rev[4] = {0, 1, 0, 1};
  const int Kd7 = 32 * 9;   // 288
  for (int k = 0; k < 4; ++k) {
    launch_conv(stream, o3, d7w + (size_t)(96 * k) * Kd7, d7b + 96 * k,
                nullptr, gates, wT, BATCH, 32, 96, 256, 256, 3, 0);
    norm_gates_kernel<<<gblocks((size_t)BATCH * NF * plane256), 256, 0, stream>>>(gates);
    propagate_kernel<<<dim3(NF, BATCH), 256, 0, stream>>>(
        Xb, gates, maxacc, dirs_hor[k], dirs_rev[k], k == 0 ? 1 : 0);
  }

  // ------------------ final projection conv ------------------
  launch_conv(stream, maxacc, pw, pb, nullptr, (float*)d_out, wT,
              BATCH, 32, 3, 256, 256, 3, 0);
}